// LinearTransformer_85341000172338
// MI455X (gfx1250) — compile-verified
//
#include <hip/hip_runtime.h>
#include <hip/hip_bf16.h>
#include <math.h>

typedef float v2f __attribute__((ext_vector_type(2)));
typedef float v8f __attribute__((ext_vector_type(8)));

// ---------------------------------------------------------------------------
// WMMA f32 GEMM:  C[m, n] = sum_k A[m,k] * W[n,k] + bias[n]   (optional ReLU)
// A: (M,K) row-major.  W: (N,K) row-major.  C: row-major with leading dim ldc.
// One wave computes one 16x16 tile via V_WMMA_F32_16X16X4_F32 over K.
//
// No load-side predication: out-of-range rows/cols clamp their row pointer to
// row 0 and compute garbage that is discarded by the guarded store (C[m,n] is
// row-local in A and column-local in W). EXEC stays all-ones for WMMA.
// ---------------------------------------------------------------------------
__global__ __launch_bounds__(128)
void wmma_gemm_kernel(const float* __restrict__ A,
                      const float* __restrict__ W,
                      const float* __restrict__ bias,
                      float* __restrict__ C,
                      int M, int N, int K, int ldc, int relu)
{
    const int lane = threadIdx.x & 31;
    const int wave = threadIdx.x >> 5;
    const int nt   = blockIdx.x * 4 + wave;   // n-tile handled by this wave
    const int mt   = blockIdx.y;              // m-tile
    const int n0   = nt * 16;
    const int m0   = mt * 16;
    if (n0 >= N) return;                      // uniform per-wave exit

    const int r  = lane & 15;
    const int kh = (lane >> 4) << 1;          // 0 or 2 (K sub-offset per half-wave)
    const int am = m0 + r;
    const int wn = n0 + r;
    // clamped row pointers: always valid addresses, dead lanes produce
    // duplicate values that are never stored
    const float* Arow = A + (size_t)(am < M ? am : 0) * K + kh;
    const float* Wrow = W + (size_t)(wn < N ? wn : 0) * K + kh;

    const int   ncol = n0 + (lane & 15);
    const float bv   = bias[ncol < N ? ncol : 0];

    v8f c;
#pragma unroll
    for (int j = 0; j < 8; ++j) c[j] = bv;

    const int Kmain = K & ~63;
    for (int k0 = 0; k0 < Kmain; k0 += 64) {
        __builtin_prefetch(Wrow + k0 + 64, 0, 0);   // uniform, EXEC untouched
#pragma unroll
        for (int kk = 0; kk < 64; kk += 4) {
            v2f a = *(const v2f*)(Arow + k0 + kk);  // global_load_b64
            v2f b = *(const v2f*)(Wrow + k0 + kk);
            c = __builtin_amdgcn_wmma_f32_16x16x4_f32(
                    false, a, false, b, (short)0, c, false, false);
        }
    }
    for (int k = Kmain; k < K; k += 4) {            // generic tail (unused here)
        v2f a = *(const v2f*)(Arow + k);
        v2f b = *(const v2f*)(Wrow + k);
        c = __builtin_amdgcn_wmma_f32_16x16x4_f32(
                false, a, false, b, (short)0, c, false, false);
    }

    const int mbase = m0 + ((lane >> 4) << 3);      // lanes 16-31 -> rows +8
#pragma unroll
    for (int j = 0; j < 8; ++j) {
        int   m = mbase + j;
        float v = c[j];
        if (relu) v = v > 0.0f ? v : 0.0f;
        if (m < M && ncol < N) C[(size_t)m * ldc + ncol] = v;
    }
}

// ---------------------------------------------------------------------------
__global__ void zero_kernel(float* p, long n)
{
    long i = (long)blockIdx.x * blockDim.x + threadIdx.x;
    if (i < n) p[i] = 0.0f;
}

// sinusoidal positional encoding added in-place to h (B,S,E)
__global__ void add_pe_kernel(float* __restrict__ h, int B, int S, int E)
{
    long i = (long)blockIdx.x * blockDim.x + threadIdx.x;
    long n = (long)B * S * E;
    if (i >= n) return;
    int e = (int)(i % E);
    int s = (int)((i / E) % S);
    float expo = (float)(e & ~1) * (-logf(10000.0f) / (float)E);
    float ang  = (float)s * __expf(expo);
    h[i] += (e & 1) ? cosf(ang) : sinf(ang);
}

// xcur[b,e] = h[b,t,e]
__global__ void gather_step_kernel(const float* __restrict__ h,
                                   float* __restrict__ xcur,
                                   int S, int E, int t, int n)
{
    int i = blockIdx.x * blockDim.x + threadIdx.x;
    if (i >= n) return;
    int b = i / E, e = i % E;
    xcur[i] = h[((size_t)b * S + t) * E + e];
}

// ---------------------------------------------------------------------------
// Linear-attention recurrent state update for one step, one layer.
// grid = (B, H), block = D (=64) threads; thread tid owns column tid of S.
//   q,k = elu(.)+1 ; optional length mask ; S += k (x) v ; z += k
//   attn[m] = (sum_d q[d]*S[d][m]) / (sum_d q[d]*z[d] + eps)
// ---------------------------------------------------------------------------
__global__ __launch_bounds__(64)
void attn_step_kernel(const float* __restrict__ qkv,
                      float* __restrict__ Sl,       // (B,H,D,D) for this layer
                      float* __restrict__ zl,       // (B,H,D)
                      float* __restrict__ attn,     // (B,E)
                      const int* __restrict__ lengths,
                      int t, int use_mask, int E, int D)
{
    const int b   = blockIdx.x;
    const int hh  = blockIdx.y;
    const int H   = gridDim.y;
    const int tid = threadIdx.x;

    __shared__ float qs[64], ks[64], vs[64], red[64];

    const float* base = qkv + (size_t)b * 3 * E + hh * D;
    float qv = base[tid];
    float kv = base[E + tid];
    float vv = base[2 * E + tid];
    qv = qv > 0.0f ? qv + 1.0f : __expf(qv);   // elu(x)+1
    kv = kv > 0.0f ? kv + 1.0f : __expf(kv);
    if (use_mask) {
        float m = (t < lengths[b]) ? 1.0f : 0.0f;
        kv *= m; vv *= m;
    }
    qs[tid] = qv; ks[tid] = kv; vs[tid] = vv;

    float* z  = zl + ((size_t)b * H + hh) * D;
    float  zn = z[tid] + kv;
    z[tid] = zn;
    red[tid] = qv * zn;
    __syncthreads();
    for (int st = 32; st > 0; st >>= 1) {
        if (tid < st) red[tid] += red[tid + st];
        __syncthreads();
    }
    const float den = red[0] + 1e-6f;

    float* Sb  = Sl + (((size_t)b * H + hh) * D) * D;
    const float vm = vs[tid];
    float num = 0.0f;
    for (int d = 0; d < D; ++d) {
        float sv = Sb[(size_t)d * D + tid] + ks[d] * vm;
        Sb[(size_t)d * D + tid] = sv;
        num += qs[d] * sv;
    }
    attn[(size_t)b * E + hh * D + tid] = num / den;
}

// ---------------------------------------------------------------------------
// xout = LayerNorm(xin + yin) * g + beta      (one block per batch row)
// ---------------------------------------------------------------------------
__global__ __launch_bounds__(256)
void residual_ln_kernel(const float* __restrict__ xin,
                        const float* __restrict__ yin,
                        const float* __restrict__ g,
                        const float* __restrict__ beta,
                        float* __restrict__ xout, int E)
{
    const int b = blockIdx.x, tid = threadIdx.x, T = blockDim.x;
    __shared__ float red[256];
    const int per = E / T;                    // E=1024, T=256 -> 4
    float v[8];
    float s = 0.0f;
    for (int i = 0; i < per; ++i) {
        int e = tid + i * T;
        v[i] = xin[(size_t)b * E + e] + yin[(size_t)b * E + e];
        s += v[i];
    }
    red[tid] = s; __syncthreads();
    for (int st = T >> 1; st > 0; st >>= 1) {
        if (tid < st) red[tid] += red[tid + st];
        __syncthreads();
    }
    const float mu = red[0] / (float)E;
    __syncthreads();
    s = 0.0f;
    for (int i = 0; i < per; ++i) { float d = v[i] - mu; s += d * d; }
    red[tid] = s; __syncthreads();
    for (int st = T >> 1; st > 0; st >>= 1) {
        if (tid < st) red[tid] += red[tid + st];
        __syncthreads();
    }
    const float rs = rsqrtf(red[0] / (float)E + 1e-5f);
    for (int i = 0; i < per; ++i) {
        int e = tid + i * T;
        xout[(size_t)b * E + e] = (v[i] - mu) * rs * g[e] + beta[e];
    }
}

// ---------------------------------------------------------------------------
extern "C" void kernel_launch(void* const* d_in, const int* in_sizes, int n_in,
                              void* d_out, int out_size, void* d_ws, size_t ws_size,
                              hipStream_t stream)
{
    const float* x       = (const float*)d_in[0];
    const int*   lengths = (const int*)  d_in[1];
    const float* emb_w   = (const float*)d_in[3];
    const float* emb_b   = (const float*)d_in[4];

    const float* eqw  = (const float*)d_in[5];
    const float* eqb  = (const float*)d_in[6];
    const float* eow  = (const float*)d_in[7];
    const float* eob  = (const float*)d_in[8];
    const float* en1g = (const float*)d_in[9];
    const float* en1b = (const float*)d_in[10];
    const float* en2g = (const float*)d_in[11];
    const float* en2b = (const float*)d_in[12];
    const float* ef1w = (const float*)d_in[13];
    const float* ef1b = (const float*)d_in[14];
    const float* ef2w = (const float*)d_in[15];
    const float* ef2b = (const float*)d_in[16];

    const float* dqw  = (const float*)d_in[17];
    const float* dqb  = (const float*)d_in[18];
    const float* dow  = (const float*)d_in[19];
    const float* dob  = (const float*)d_in[20];
    const float* dn1g = (const float*)d_in[21];
    const float* dn1b = (const float*)d_in[22];
    const float* dn2g = (const float*)d_in[23];
    const float* dn2b = (const float*)d_in[24];
    const float* df1w = (const float*)d_in[25];
    const float* df1b = (const float*)d_in[26];
    const float* df2w = (const float*)d_in[27];
    const float* df2b = (const float*)d_in[28];

    const float* fc1_w = (const float*)d_in[29];
    const float* fc1_b = (const float*)d_in[30];
    const float* fc2_w = (const float*)d_in[31];
    const float* fc2_b = (const float*)d_in[32];

    // derive dims from input sizes (deterministic; no device reads)
    const int B    = in_sizes[1];
    const int E    = in_sizes[4];
    const int IN   = in_sizes[3] / E;
    const int S    = in_sizes[0] / (B * IN);
    const int L    = in_sizes[6] / (3 * E);
    const int C    = in_sizes[32];
    const int TOUT = out_size / (B * C);
    const int H    = 16;                 // NHEAD
    const int D    = E / H;              // 64

    // workspace carve-up (floats)
    float* ws   = (float*)d_ws;
    float* hbuf = ws;  ws += (size_t)B * S * E;           // embedded sequence
    float* Sst  = ws;  ws += (size_t)L * B * H * D * D;   // recurrent S state
    float* zst  = ws;  ws += (size_t)L * B * H * D;       // recurrent z state
    float* xcur = ws;  ws += (size_t)B * E;
    float* qkvb = ws;  ws += (size_t)B * 3 * E;
    float* attb = ws;  ws += (size_t)B * E;
    float* ybuf = ws;  ws += (size_t)B * E;
    float* ffb  = ws;  ws += (size_t)B * E;

    auto gemm = [&](const float* A, const float* W, const float* bias, float* Cp,
                    int M, int N, int K, int ldc, int relu) {
        int ntile = (N + 15) / 16;
        dim3 grid((ntile + 3) / 4, (M + 15) / 16);
        wmma_gemm_kernel<<<grid, 128, 0, stream>>>(A, W, bias, Cp, M, N, K, ldc, relu);
    };

    // zero recurrent state (Sst and zst are contiguous)
    {
        long n = (long)L * B * H * D * D + (long)L * B * H * D;
        zero_kernel<<<(unsigned)((n + 255) / 256), 256, 0, stream>>>(Sst, n);
    }

    // h = x @ emb_w^T + emb_b ; h += sinusoidal PE
    gemm(x, emb_w, emb_b, hbuf, B * S, E, IN, E, 0);
    {
        long n = (long)B * S * E;
        add_pe_kernel<<<(unsigned)((n + 255) / 256), 256, 0, stream>>>(hbuf, B, S, E);
    }

    auto layer = [&](const float* qw, const float* qb, const float* ow, const float* ob,
                     const float* g1, const float* b1, const float* g2, const float* b2,
                     const float* f1w, const float* f1b, const float* f2w, const float* f2b,
                     float* Sl, float* zl, int t, int use_mask) {
        gemm(xcur, qw, qb, qkvb, B, 3 * E, E, 3 * E, 0);
        attn_step_kernel<<<dim3(B, H), D, 0, stream>>>(qkvb, Sl, zl, attb,
                                                       lengths, t, use_mask, E, D);
        gemm(attb, ow, ob, ybuf, B, E, E, E, 0);
        residual_ln_kernel<<<B, 256, 0, stream>>>(xcur, ybuf, g1, b1, xcur, E);
        gemm(xcur, f1w, f1b, ffb, B, E, E, E, 1);     // ReLU fused
        gemm(ffb,  f2w, f2b, ybuf, B, E, E, E, 0);
        residual_ln_kernel<<<B, 256, 0, stream>>>(xcur, ybuf, g2, b2, xcur, E);
    };

    // ---------------- encoder scan over S time steps ----------------
    for (int t = 0; t < S; ++t) {
        int n = B * E;
        gather_step_kernel<<<(n + 255) / 256, 256, 0, stream>>>(hbuf, xcur, S, E, t, n);
        for (int l = 0; l < L; ++l) {
            layer(eqw  + (size_t)l * 3 * E * E, eqb  + (size_t)l * 3 * E,
                  eow  + (size_t)l * E * E,     eob  + (size_t)l * E,
                  en1g + (size_t)l * E,         en1b + (size_t)l * E,
                  en2g + (size_t)l * E,         en2b + (size_t)l * E,
                  ef1w + (size_t)l * E * E,     ef1b + (size_t)l * E,
                  ef2w + (size_t)l * E * E,     ef2b + (size_t)l * E,
                  Sst + (size_t)l * B * H * D * D,
                  zst + (size_t)l * B * H * D, t, /*mask*/1);
        }
    }

    // ---------------- decoder: TOUT autoregressive steps ----------------
    float* out = (float*)d_out;                  // (B, TOUT, C)
    for (int t = 0; t < TOUT; ++t) {
        for (int l = 0; l < L; ++l) {
            layer(dqw  + (size_t)l * 3 * E * E, dqb  + (size_t)l * 3 * E,
                  dow  + (size_t)l * E * E,     dob  + (size_t)l * E,
                  dn1g + (size_t)l * E,         dn1b + (size_t)l * E,
                  dn2g + (size_t)l * E,         dn2b + (size_t)l * E,
                  df1w + (size_t)l * E * E,     df1b + (size_t)l * E,
                  df2w + (size_t)l * E * E,     df2b + (size_t)l * E,
                  Sst + (size_t)l * B * H * D * D,
                  zst + (size_t)l * B * H * D, 0, /*mask*/0);
        }
        gemm(xcur, fc1_w, fc1_b, ffb, B, E, E, E, 1);
        gemm(ffb,  fc2_w, fc2_b, out + (size_t)t * C, B, C, E, TOUT * C, 0);
    }
}